// AutoencoderModel_65481071400619
// MI455X (gfx1250) — compile-verified
//
#include <hip/hip_runtime.h>
#include <hip/hip_bf16.h>

// ---------------------------------------------------------------------------
// 21-qubit state-vector simulator, CDNA5 (gfx1250) WMMA-fused.
//   - 4-qubit fused gates applied as 16x16 complex GEMM via V_WMMA_F32_16X16X4_F32
//   - full CNOT chain = one Gray-code gather (out[j] = in[j ^ (j>>1)])
//   - wire-20 2x2 gate fused into the gather pass
// State kept in d_ws (two 16MB ping-pong buffers), everything L2-resident.
// ---------------------------------------------------------------------------

#define NQ 21
#define NSTATE (1u << NQ)          // 2^21 amplitudes
#define NCTX   (1u << (NQ - 4))   // 2^17 contexts per 4-qubit pass
#define NTILES (NCTX / 16)        // 8192 column tiles of 16

typedef __attribute__((ext_vector_type(2))) float v2f;
typedef __attribute__((ext_vector_type(8))) float v8f;

__device__ __forceinline__ float2 cmul(float2 a, float2 b) {
  return make_float2(a.x * b.x - a.y * b.y, a.x * b.y + a.y * b.x);
}

// ---------------------------------------------------------------------------
// Build fused 16x16 group matrices (Kron of four 2x2 gates) + wire-20 2x2.
// Blocks 0..19: matrix p = layer*5+g (layer 0 = RY encoding, 1..3 = Rot reps)
// Blocks 20..22: u2[r] = Rot(r,20) (rep 0 also folds in RY(x[20])).
// ---------------------------------------------------------------------------
__device__ __forceinline__ void rot_gate(const float* params, int r, int w,
                                         float2 G[2][2]) {
  float phi = params[(r * NQ + w) * 3 + 0];
  float th  = params[(r * NQ + w) * 3 + 1];
  float om  = params[(r * NQ + w) * 3 + 2];
  float ch = cosf(0.5f * th), sh = sinf(0.5f * th);
  float ap = -0.5f * (phi + om), am = 0.5f * (phi - om);
  G[0][0] = make_float2(cosf(ap) * ch,  sinf(ap) * ch);
  G[0][1] = make_float2(-cosf(am) * sh, -sinf(am) * sh);
  G[1][0] = make_float2(cosf(am) * sh, -sinf(am) * sh);
  G[1][1] = make_float2(cosf(ap) * ch, -sinf(ap) * ch);
}

__global__ void build_mats(const float* __restrict__ x,
                           const float* __restrict__ params,
                           float* __restrict__ mats,   // 20 * 512 floats
                           float* __restrict__ u2s) {  // 3 * 8 floats
  int p = blockIdx.x;
  if (p < 20) {
    int layer = p / 5, g = p % 5;
    float2 G[4][2][2];
    for (int j = 0; j < 4; ++j) {
      int w = 4 * g + j;
      if (layer == 0) {
        float c = cosf(0.5f * x[w]), s = sinf(0.5f * x[w]);
        G[j][0][0] = make_float2(c, 0.f);  G[j][0][1] = make_float2(-s, 0.f);
        G[j][1][0] = make_float2(s, 0.f);  G[j][1][1] = make_float2(c, 0.f);
      } else {
        rot_gate(params, layer - 1, w, G[j]);
      }
    }
    int tid = threadIdx.x;  // 256 threads: one matrix entry each
    int mp = tid >> 4, m = tid & 15;
    float2 prod = make_float2(1.f, 0.f);
    for (int j = 0; j < 4; ++j) {
      int bp = (mp >> (3 - j)) & 1, bm = (m >> (3 - j)) & 1;
      prod = cmul(prod, G[j][bp][bm]);
    }
    mats[p * 512 + tid]       = prod.x;  // Ur row-major
    mats[p * 512 + 256 + tid] = prod.y;  // Ui row-major
  } else if (p < 23 && threadIdx.x == 0) {
    int r = p - 20;
    float2 R[2][2];
    rot_gate(params, r, 20, R);
    float2 M[2][2];
    if (r == 0) {  // fold encoding RY(x20): state' = Rot * (RY * state)
      float c = cosf(0.5f * x[20]), s = sinf(0.5f * x[20]);
      float2 Y[2][2] = {{make_float2(c, 0.f), make_float2(-s, 0.f)},
                        {make_float2(s, 0.f), make_float2(c, 0.f)}};
      for (int a = 0; a < 2; ++a)
        for (int b = 0; b < 2; ++b) {
          float2 acc = make_float2(0.f, 0.f);
          for (int k = 0; k < 2; ++k) {
            float2 t = cmul(R[a][k], Y[k][b]);
            acc.x += t.x; acc.y += t.y;
          }
          M[a][b] = acc;
        }
    } else {
      for (int a = 0; a < 2; ++a)
        for (int b = 0; b < 2; ++b) M[a][b] = R[a][b];
    }
    float* o = u2s + r * 8;
    o[0] = M[0][0].x; o[1] = M[0][0].y; o[2] = M[0][1].x; o[3] = M[0][1].y;
    o[4] = M[1][0].x; o[5] = M[1][0].y; o[6] = M[1][1].x; o[7] = M[1][1].y;
  }
}

// ---------------------------------------------------------------------------
// |0...0> initialization
// ---------------------------------------------------------------------------
__global__ void init_state(float2* __restrict__ st) {
  unsigned i = blockIdx.x * blockDim.x + threadIdx.x;
  st[i] = make_float2(i == 0 ? 1.f : 0.f, 0.f);
}

// ---------------------------------------------------------------------------
// Apply 16x16 complex gate on 4 contiguous bits [s+3..s], in place.
// Per wave: load U into A-operands once, then stream 16-column tiles.
// GEMM: Out = U * In ; complex via 4 real WMMAs per K-chunk (K=16 -> 4 chunks).
//   A 16x4 layout: lane holds row M=lane&15; lane<16 -> K{0,1}, lane>=16 -> K{2,3}
//   B 4x16 layout: lane holds col N=lane&15; same K-half split
//   D 16x16 layout: lane holds col N=lane&15; lane<16 rows 0-7, lane>=16 rows 8-15
// ---------------------------------------------------------------------------
__global__ void gate4_pass(float2* __restrict__ st,
                           const float* __restrict__ U, int s) {
  const float* Ur = U;
  const float* Ui = U + 256;
  unsigned lane   = threadIdx.x & 31u;
  unsigned wave   = (blockIdx.x * blockDim.x + threadIdx.x) >> 5;
  unsigned nwaves = (gridDim.x * blockDim.x) >> 5;
  unsigned row    = lane & 15u;
  unsigned khalf  = (lane >> 4) << 1;  // 0 or 2

  v2f ar[4], ai[4], an[4];
  for (int kk = 0; kk < 4; ++kk) {
    unsigned kb = 4u * kk + khalf;
    ar[kk] = (v2f){Ur[row * 16 + kb], Ur[row * 16 + kb + 1]};
    ai[kk] = (v2f){Ui[row * 16 + kb], Ui[row * 16 + kb + 1]};
    an[kk] = (v2f){-ai[kk][0], -ai[kk][1]};
  }

  unsigned lowmask = (1u << s) - 1u;
  for (unsigned t = wave; t < NTILES; t += nwaves) {
    unsigned ctx  = t * 16u + row;          // this lane's column context
    unsigned l    = ctx & lowmask;
    unsigned h    = ctx >> s;
    unsigned base = (h << (s + 4)) + l;

    v2f br[4], bi[4];
    for (int kk = 0; kk < 4; ++kk) {
      unsigned kb = 4u * kk + khalf;
      float2 a0 = st[base + (kb << s)];
      float2 a1 = st[base + ((kb + 1u) << s)];
      br[kk] = (v2f){a0.x, a1.x};
      bi[kk] = (v2f){a0.y, a1.y};
    }

    v8f dr = {0.f, 0.f, 0.f, 0.f, 0.f, 0.f, 0.f, 0.f};
    v8f di = {0.f, 0.f, 0.f, 0.f, 0.f, 0.f, 0.f, 0.f};
    for (int kk = 0; kk < 4; ++kk) {
      // Re += Ur*Sr - Ui*Si ; Im += Ur*Si + Ui*Sr   (fp32 WMMA, full precision)
      dr = __builtin_amdgcn_wmma_f32_16x16x4_f32(false, ar[kk], false, br[kk],
                                                 (short)0, dr, false, false);
      dr = __builtin_amdgcn_wmma_f32_16x16x4_f32(false, an[kk], false, bi[kk],
                                                 (short)0, dr, false, false);
      di = __builtin_amdgcn_wmma_f32_16x16x4_f32(false, ar[kk], false, bi[kk],
                                                 (short)0, di, false, false);
      di = __builtin_amdgcn_wmma_f32_16x16x4_f32(false, ai[kk], false, br[kk],
                                                 (short)0, di, false, false);
    }

    unsigned rbase = (lane >> 4) * 8u;  // rows 0-7 or 8-15
    for (int v = 0; v < 8; ++v) {
      unsigned r_ = rbase + (unsigned)v;
      st[base + (r_ << s)] = make_float2(dr[v], di[v]);
    }
  }
}

// ---------------------------------------------------------------------------
// Whole CNOT chain (20 gates) + wire-20 2x2 gate, as one gather:
//   out[j] = sum_b u2[g(j)&1][b] * in[(g(j)&~1)|b],  g(j) = j ^ (j>>1)
// Thread handles output pair {2t, 2t+1} which shares one input pair.
// ---------------------------------------------------------------------------
__global__ void chain_pass(const float2* __restrict__ in,
                           float2* __restrict__ out,
                           const float* __restrict__ u2) {
  unsigned t  = blockIdx.x * blockDim.x + threadIdx.x;  // 2^20 threads
  unsigned jb = t << 1;
  unsigned g  = jb ^ (jb >> 1);
  unsigned b  = g & ~1u;
  float2 a0 = in[b], a1 = in[b | 1u];
  float r0re = u2[0] * a0.x - u2[1] * a0.y + u2[2] * a1.x - u2[3] * a1.y;
  float r0im = u2[0] * a0.y + u2[1] * a0.x + u2[2] * a1.y + u2[3] * a1.x;
  float r1re = u2[4] * a0.x - u2[5] * a0.y + u2[6] * a1.x - u2[7] * a1.y;
  float r1im = u2[4] * a0.y + u2[5] * a0.x + u2[6] * a1.y + u2[7] * a1.x;
  unsigned r = g & 1u;
  out[jb]      = r ? make_float2(r1re, r1im) : make_float2(r0re, r0im);
  out[jb | 1u] = r ? make_float2(r0re, r0im) : make_float2(r1re, r1im);
}

// ---------------------------------------------------------------------------
// <Z_w> partial sums (deterministic two-stage reduction, no float atomics)
// ---------------------------------------------------------------------------
__global__ void expect_partial(const float2* __restrict__ st,
                               float* __restrict__ partials) {
  __shared__ float red[256];
  float acc[NQ];
  for (int w = 0; w < NQ; ++w) acc[w] = 0.f;
  unsigned tid = threadIdx.x;
  unsigned gtid = blockIdx.x * blockDim.x + tid;
  unsigned stride = gridDim.x * blockDim.x;
  for (unsigned i = gtid; i < NSTATE; i += stride) {
    float2 a = st[i];
    float p = a.x * a.x + a.y * a.y;
    for (int w = 0; w < NQ; ++w) {
      acc[w] += ((i >> (NQ - 1 - w)) & 1u) ? -p : p;
    }
  }
  for (int w = 0; w < NQ; ++w) {
    red[tid] = acc[w];
    __syncthreads();
    for (int o = 128; o > 0; o >>= 1) {
      if ((int)tid < o) red[tid] += red[tid + o];
      __syncthreads();
    }
    if (tid == 0) partials[blockIdx.x * NQ + w] = red[0];
    __syncthreads();
  }
}

__global__ void expect_final(const float* __restrict__ partials,
                             float* __restrict__ out, int nblocks) {
  int w = threadIdx.x;
  if (w < NQ) {
    float s = 0.f;
    for (int b = 0; b < nblocks; ++b) s += partials[b * NQ + w];
    out[w] = s;
  }
}

// ---------------------------------------------------------------------------
extern "C" void kernel_launch(void* const* d_in, const int* in_sizes, int n_in,
                              void* d_out, int out_size, void* d_ws, size_t ws_size,
                              hipStream_t stream) {
  (void)in_sizes; (void)n_in; (void)out_size; (void)ws_size;
  const float* x      = (const float*)d_in[0];   // (21,) f32
  const float* params = (const float*)d_in[1];   // (3,21,3) f32

  char* ws = (char*)d_ws;
  float2* stA = (float2*)ws;                          // 16 MB
  float2* stB = (float2*)(ws + (size_t)(16u << 20));  // 16 MB
  float*  mats = (float*)(ws + (size_t)(32u << 20));  // 20*512 floats
  float*  u2s  = mats + 20 * 512;                     // 3*8 floats
  float*  partials = u2s + 24;                        // 256*21 floats

  build_mats<<<23, 256, 0, stream>>>(x, params, mats, u2s);
  init_state<<<NSTATE / 256, 256, 0, stream>>>(stA);

  // encoding layer (real RY Kron matrices; Ui == 0)
  for (int g = 0; g < 5; ++g)
    gate4_pass<<<128, 256, 0, stream>>>(stA, mats + g * 512, 17 - 4 * g);

  float2* cur = stA;
  float2* oth = stB;
  for (int r = 0; r < 3; ++r) {
    for (int g = 0; g < 5; ++g)
      gate4_pass<<<128, 256, 0, stream>>>(cur, mats + ((r + 1) * 5 + g) * 512,
                                          17 - 4 * g);
    chain_pass<<<(NSTATE / 2) / 256, 256, 0, stream>>>(cur, oth, u2s + r * 8);
    float2* tmp = cur; cur = oth; oth = tmp;
  }

  expect_partial<<<256, 256, 0, stream>>>(cur, partials);
  expect_final<<<1, 32, 0, stream>>>(partials, (float*)d_out, 256);
}